// PureFEPLayer_40973988004474
// MI455X (gfx1250) — compile-verified
//
#include <hip/hip_runtime.h>
#include <math.h>

#define K_DIM 255
#define KP    256
#define N_SEQ 512
#define B_SZ  2
#define V_SZ  8000
#define NT16  (N_SEQ/16)
#define BN    (B_SZ*N_SEQ)
#define EPS_F      1e-6f
#define ALPHA_F    0.1f
#define LAMBDA_B_F 1.0f
#define LAMBDA_O_F 1.0f
#define KAPPA_F    0.1f
#define MU_LR_F    0.1f
#define SIG_LR_F   0.025f
#define CLIP_F     1.0f

typedef __attribute__((ext_vector_type(2))) float v2f;
typedef __attribute__((ext_vector_type(8))) float v8f;

__device__ __forceinline__ v8f zero8() {
  v8f z = {0.f,0.f,0.f,0.f,0.f,0.f,0.f,0.f};
  return z;
}

// D = A(16x4,f32) * B(4x16,f32) + C(16x16,f32) on the matrix pipe (exact f32).
__device__ __forceinline__ v8f wmma4(v2f a, v2f b, v8f c) {
  return __builtin_amdgcn_wmma_f32_16x16x4_f32(false, a, false, b, (short)0, c,
                                               false, false);
}

// Wave-local LDS store->load ordering (E tile is written and re-read by the
// same wave between WMMA stages; DS ops from one wave are in-order, the
// explicit wait makes the cross-lane dependency airtight).
__device__ __forceinline__ void wave_lds_fence() {
  asm volatile("s_wait_dscnt 0" ::: "memory");
}

// ---------------------------------------------------------------- init state
__global__ void k_init(const float* __restrict__ x, float* __restrict__ mu,
                       float* __restrict__ sig) {
  int row = blockIdx.x, t = threadIdx.x;
  size_t base = (size_t)row * KP;
  mu[base + t]  = (t < K_DIM) ? x[(size_t)row * K_DIM + t] : 0.f;  // zero pad
  sig[base + t] = 0.1f;
}

// --------------------------------------------------- per-step derived arrays
// U = 1/sq, W2 = mu*U, Y1 = sq + mu^2 (all zero in the pad column),
// cvec = sum(mu^2*U) - K - logdet, dvec = logdet.
__global__ void k_prep(const float* __restrict__ mu, const float* __restrict__ sig,
                       float* __restrict__ U, float* __restrict__ W2,
                       float* __restrict__ Y1, float* __restrict__ cvec,
                       float* __restrict__ dvec) {
  __shared__ float redA[256], redB[256];
  int row = blockIdx.x, t = threadIdx.x;
  size_t base = (size_t)row * KP;
  float ld = 0.f, m2 = 0.f;
  if (t < K_DIM) {
    float sq = fmaxf(sig[base + t], EPS_F);
    float u  = 1.f / sq;
    float m  = mu[base + t];
    U[base + t]  = u;
    W2[base + t] = m * u;
    Y1[base + t] = sq + m * m;
    ld = __logf(sq);
    m2 = m * m * u;
  } else {
    U[base + t] = 0.f; W2[base + t] = 0.f; Y1[base + t] = 0.f;
  }
  redA[t] = ld; redB[t] = m2;
  __syncthreads();
  for (int s = 128; s > 0; s >>= 1) {
    if (t < s) { redA[t] += redA[t + s]; redB[t] += redB[t + s]; }
    __syncthreads();
  }
  if (t == 0) {
    dvec[row] = redA[0];
    cvec[row] = redB[0] - (float)K_DIM - redA[0];
  }
}

// -------------------------------------------- causal KL attention (flash-ish)
// One block = 16 query rows. 4 waves each rebuild the 16x16 exp(-kappa*kl)
// tile (scores <= 0, max fixed at 0 -> no rescaling) and apply it to their
// private 64-column K slice with register v8f accumulators.
__global__ void __launch_bounds__(128) k_attn(
    const float* __restrict__ MU, const float* __restrict__ U,
    const float* __restrict__ W2, const float* __restrict__ Y1,
    const float* __restrict__ cvec, const float* __restrict__ dvec,
    float* __restrict__ bm, float* __restrict__ bY) {
  __shared__ float Ui[16 * KP];
  __shared__ float Wi[16 * KP];
  __shared__ float El[4 * 16 * 16];
  __shared__ float srow[16];
  int b = blockIdx.x / NT16, it = blockIdx.x % NT16;
  int i0 = it * 16, bN = b * N_SEQ;
  int tid = threadIdx.x, wv = tid >> 5, lane = tid & 31;
  int half = lane >> 4, col = lane & 15;

  for (int idx = tid; idx < 16 * KP; idx += 128) {
    int r = idx >> 8, k = idx & (KP - 1);
    Ui[idx] = U[(size_t)(bN + i0 + r) * KP + k];
    Wi[idx] = W2[(size_t)(bN + i0 + r) * KP + k];
  }
  if (tid < 16) srow[tid] = 0.f;
  __syncthreads();

  float cvi[8], sacc[8];
  v8f accM[4], accY[4];
#pragma unroll
  for (int r = 0; r < 8; ++r) {
    cvi[r] = cvec[bN + i0 + r + (half << 3)];
    sacc[r] = 0.f;
  }
#pragma unroll
  for (int kt = 0; kt < 4; ++kt) { accM[kt] = zero8(); accY[kt] = zero8(); }
  int kbase = wv * 64;

  for (int jt = 0; jt <= it; ++jt) {
    int j0 = jt * 16;
    // --- scores: s1 = u_i . (sq+mu^2)_j ; s2 = (mu u)_i . mu_j
    v8f s1 = zero8(), s2 = zero8();
    const float* y1r = Y1 + (size_t)(bN + j0 + col) * KP;
    const float* mur = MU + (size_t)(bN + j0 + col) * KP;
    for (int kb = 0; kb < KP; kb += 4) {
      int k0 = kb + (half << 1);
      v2f a1, a2, b1, b2;
      a1.x = Ui[col * KP + k0]; a1.y = Ui[col * KP + k0 + 1];
      a2.x = Wi[col * KP + k0]; a2.y = Wi[col * KP + k0 + 1];
      b1.x = y1r[k0]; b1.y = y1r[k0 + 1];
      b2.x = mur[k0]; b2.y = mur[k0 + 1];
      s1 = wmma4(a1, b1, s1);
      s2 = wmma4(a2, b2, s2);
    }
    float dvj = dvec[bN + j0 + col];
#pragma unroll
    for (int r = 0; r < 8; ++r) {
      int M = r + (half << 3);
      float kl = 0.5f * (s1[r] - 2.f * s2[r] + cvi[r] + dvj);
      float e = ((j0 + col) <= (i0 + M)) ? __expf(-KAPPA_F * kl) : 0.f;
      sacc[r] += e;
      El[wv * 256 + M * 16 + col] = e;
    }
    wave_lds_fence();
    // --- apply: acc[i,k] += sum_j E[i,j] * X[j,k] for this wave's K slice
#pragma unroll
    for (int kt = 0; kt < 4; ++kt) {
      int kc = kbase + kt * 16 + col;
#pragma unroll
      for (int jj = 0; jj < 16; jj += 4) {
        v2f a, b1, b2;
        a.x = El[wv * 256 + col * 16 + jj + (half << 1)];
        a.y = El[wv * 256 + col * 16 + jj + (half << 1) + 1];
        int jr = j0 + jj + (half << 1);
        const float* m0 = MU + (size_t)(bN + jr) * KP + kc;
        const float* y0 = Y1 + (size_t)(bN + jr) * KP + kc;
        b1.x = m0[0]; b1.y = m0[KP];
        b2.x = y0[0]; b2.y = y0[KP];
        accM[kt] = wmma4(a, b1, accM[kt]);
        accY[kt] = wmma4(a, b2, accY[kt]);
      }
    }
  }
  __syncthreads();
  if (wv == 0) {
#pragma unroll
    for (int r = 0; r < 8; ++r) atomicAdd(&srow[r + (half << 3)], sacc[r]);
  }
  __syncthreads();
#pragma unroll
  for (int kt = 0; kt < 4; ++kt) {
    int kc = kbase + kt * 16 + col;
#pragma unroll
    for (int r = 0; r < 8; ++r) {
      int M = r + (half << 3);
      float inv = 1.f / srow[M];
      bm[(size_t)(bN + i0 + M) * KP + kc] = accM[kt][r] * inv;
      bY[(size_t)(bN + i0 + M) * KP + kc] = accY[kt][r] * inv;
    }
  }
}

// --------------------- vocab softmax + CE backprop, fused flash-style (V=8000)
__global__ void __launch_bounds__(128) k_obs(
    const float* __restrict__ MU, const float* __restrict__ Wg,
    const int* __restrict__ tgt, float* __restrict__ gobs) {
  __shared__ float Mi[16 * KP];
  __shared__ float El[4 * 16 * 16];
  __shared__ float wpart[4 * 32 * 8];
  __shared__ float accS[16 * KP];
  __shared__ float mrow[16];
  __shared__ float srow[16];
  int b = blockIdx.x / NT16, it = blockIdx.x % NT16;
  int i0 = it * 16, bN = b * N_SEQ;
  int tid = threadIdx.x, wv = tid >> 5, lane = tid & 31;
  int half = lane >> 4, col = lane & 15;

  for (int idx = tid; idx < 16 * KP; idx += 128) {
    int r = idx >> 8, k = idx & (KP - 1);
    Mi[idx]   = MU[(size_t)(bN + i0 + r) * KP + k];
    accS[idx] = 0.f;
  }
  __syncthreads();

  // pass 1: per-row logit maxima (waves split the 500 v-tiles, 125 each)
  float mx[8];
#pragma unroll
  for (int r = 0; r < 8; ++r) mx[r] = -1e30f;
  for (int vt = wv; vt < V_SZ / 16; vt += 4) {
    int v0 = vt * 16;
    v8f l = zero8();
    const float* wr = Wg + (size_t)(v0 + col) * K_DIM;
    for (int kb = 0; kb < KP; kb += 4) {
      int k0 = kb + (half << 1);
      v2f a, bb;
      a.x = Mi[col * KP + k0]; a.y = Mi[col * KP + k0 + 1];
      bb.x = wr[k0];                                   // k0 <= 254 always
      bb.y = (k0 + 1 < K_DIM) ? wr[k0 + 1] : 0.f;      // guard pad column
      l = wmma4(a, bb, l);
    }
#pragma unroll
    for (int r = 0; r < 8; ++r) mx[r] = fmaxf(mx[r], l[r]);
  }
#pragma unroll
  for (int r = 0; r < 8; ++r) wpart[wv * 256 + lane * 8 + r] = mx[r];
  __syncthreads();
  if (tid < 16) {
    int row = tid, lb = (row < 8) ? 0 : 16, rr = row & 7;
    float m = -1e30f;
    for (int w2 = 0; w2 < 4; ++w2)
      for (int l2 = 0; l2 < 16; ++l2)
        m = fmaxf(m, wpart[w2 * 256 + (lb + l2) * 8 + rr]);
    mrow[row] = m;
    srow[row] = 0.f;
  }
  __syncthreads();

  // pass 2: e = exp(l - max), acc[i,k] += e * W[v,k]; register K accumulators
  float ssum[8];
  v8f accO[16];
#pragma unroll
  for (int r = 0; r < 8; ++r) ssum[r] = 0.f;
#pragma unroll
  for (int kt = 0; kt < 16; ++kt) accO[kt] = zero8();
  for (int vt = wv; vt < V_SZ / 16; vt += 4) {
    int v0 = vt * 16;
    v8f l = zero8();
    const float* wr = Wg + (size_t)(v0 + col) * K_DIM;
    for (int kb = 0; kb < KP; kb += 4) {
      int k0 = kb + (half << 1);
      v2f a, bb;
      a.x = Mi[col * KP + k0]; a.y = Mi[col * KP + k0 + 1];
      bb.x = wr[k0];
      bb.y = (k0 + 1 < K_DIM) ? wr[k0 + 1] : 0.f;
      l = wmma4(a, bb, l);
    }
#pragma unroll
    for (int r = 0; r < 8; ++r) {
      int M = r + (half << 3);
      float e = __expf(l[r] - mrow[M]);
      ssum[r] += e;
      El[wv * 256 + M * 16 + col] = e;
    }
    wave_lds_fence();
#pragma unroll
    for (int kt = 0; kt < 16; ++kt) {
      int kc = kt * 16 + col;
      bool ok = kc < K_DIM;
#pragma unroll
      for (int vv = 0; vv < 16; vv += 4) {
        v2f a, bb;
        a.x = El[wv * 256 + col * 16 + vv + (half << 1)];
        a.y = El[wv * 256 + col * 16 + vv + (half << 1) + 1];
        int vr = v0 + vv + (half << 1);
        bb.x = ok ? Wg[(size_t)vr * K_DIM + kc] : 0.f;
        bb.y = ok ? Wg[(size_t)(vr + 1) * K_DIM + kc] : 0.f;
        accO[kt] = wmma4(a, bb, accO[kt]);
      }
    }
  }
  __syncthreads();
#pragma unroll
  for (int r = 0; r < 8; ++r) atomicAdd(&srow[r + (half << 3)], ssum[r]);
#pragma unroll
  for (int kt = 0; kt < 16; ++kt) {
    int kc = kt * 16 + col;
#pragma unroll
    for (int r = 0; r < 8; ++r)
      atomicAdd(&accS[(r + (half << 3)) * KP + kc], accO[kt][r]);
  }
  __syncthreads();
  const float coef = LAMBDA_O_F / (float)(B_SZ * N_SEQ);
  for (int idx = tid; idx < 16 * KP; idx += 128) {
    int r = idx >> 8, k = idx & (KP - 1);
    float val = 0.f;
    if (k < K_DIM) {
      int tg = tgt[bN + i0 + r];
      val = coef * (accS[idx] / srow[r] - Wg[(size_t)tg * K_DIM + k]);
    }
    gobs[(size_t)(bN + i0 + r) * KP + k] = val;
  }
}

// -------------------------------------------- gradients + natural grads + norm
__global__ void k_gradnat(
    const float* __restrict__ mu, const float* __restrict__ sig,
    const float* __restrict__ bm, const float* __restrict__ bY,
    const float* __restrict__ gobs, const float* __restrict__ pmu,
    const float* __restrict__ psig, float* __restrict__ natmu,
    float* __restrict__ natsig, float* __restrict__ gsum) {
  __shared__ float red[256];
  int row = blockIdx.x, t = threadIdx.x;
  int n = row % N_SEQ;
  size_t base = (size_t)row * KP;
  float acc = 0.f;
  if (t < K_DIM) {
    float m = mu[base + t];
    float sq = fmaxf(sig[base + t], EPS_F), u = 1.f / sq;
    float sp = fmaxf(psig[(size_t)n * K_DIM + t], EPS_F);
    float pm = pmu[(size_t)n * K_DIM + t];
    float bmv = bm[base + t], bYv = bY[base + t];
    float gm = ALPHA_F * (m - pm) / sp + LAMBDA_B_F * (m - bmv) * u + gobs[base + t];
    float bs_m2 = bYv + m * m - 2.f * m * bmv;  // beta.(sq+mu^2) + mu^2 - 2 mu bm
    float gs = ALPHA_F * 0.5f * (1.f / sp - u)
             + LAMBDA_B_F * 0.5f * (u - bs_m2 * u * u);
    float nm = gm * sq, ns = 2.f * gs * sq * sq;
    natmu[base + t] = nm;
    natsig[base + t] = ns;
    acc = nm * nm;
  }
  red[t] = acc;
  __syncthreads();
  for (int s = 128; s > 0; s >>= 1) {
    if (t < s) red[t] += red[t + s];
    __syncthreads();
  }
  if (t == 0) atomicAdd(gsum, red[0]);
}

__global__ void k_zero(float* g) {
  if (threadIdx.x == 0 && blockIdx.x == 0) g[0] = 0.f;
}

__global__ void k_update(float* __restrict__ mu, float* __restrict__ sig,
                         const float* __restrict__ natmu,
                         const float* __restrict__ natsig,
                         const float* __restrict__ gsum) {
  int row = blockIdx.x, t = threadIdx.x;
  if (t >= K_DIM) return;                 // pad column stays zero in mu
  size_t base = (size_t)row * KP;
  float scale = fminf(1.f, CLIP_F / (sqrtf(gsum[0]) + EPS_F));
  mu[base + t] = mu[base + t] - MU_LR_F * scale * natmu[base + t];
  float sq = fmaxf(sig[base + t], EPS_F);
  sig[base + t] = fmaxf(sq - SIG_LR_F * scale * natsig[base + t], EPS_F);
}

// --------------------------------------------------- final logits (B,N,V) out
__global__ void __launch_bounds__(128) k_logits(
    const float* __restrict__ MU, const float* __restrict__ Wg,
    float* __restrict__ out) {
  __shared__ float Mi[16 * KP];
  int b = blockIdx.x / NT16, it = blockIdx.x % NT16;
  int i0 = it * 16, bN = b * N_SEQ;
  int tid = threadIdx.x, wv = tid >> 5, lane = tid & 31;
  int half = lane >> 4, col = lane & 15;
  for (int idx = tid; idx < 16 * KP; idx += 128) {
    int r = idx >> 8, k = idx & (KP - 1);
    Mi[idx] = MU[(size_t)(bN + i0 + r) * KP + k];
  }
  __syncthreads();
  int v0 = blockIdx.y * 64 + wv * 16;
  v8f l = zero8();
  const float* wr = Wg + (size_t)(v0 + col) * K_DIM;
  for (int kb = 0; kb < KP; kb += 4) {
    int k0 = kb + (half << 1);
    v2f a, bb;
    a.x = Mi[col * KP + k0]; a.y = Mi[col * KP + k0 + 1];
    bb.x = wr[k0];
    bb.y = (k0 + 1 < K_DIM) ? wr[k0 + 1] : 0.f;
    l = wmma4(a, bb, l);
  }
#pragma unroll
  for (int r = 0; r < 8; ++r) {
    int M = r + (half << 3);
    out[(size_t)(bN + i0 + M) * V_SZ + v0 + col] = l[r];
  }
}

extern "C" void kernel_launch(void* const* d_in, const int* in_sizes, int n_in,
                              void* d_out, int out_size, void* d_ws,
                              size_t ws_size, hipStream_t stream) {
  const float* x    = (const float*)d_in[0];
  const float* Wg   = (const float*)d_in[1];
  const float* pmu  = (const float*)d_in[2];
  const float* psig = (const float*)d_in[3];
  // d_in[4] = generators: unused — phi stays 0 for all 20 steps, so the gauge
  // rotation term (phi^a G_a mu) is identically zero and mu_rot == mu_q.
  const int* tgt    = (const int*)d_in[5];
  float* out = (float*)d_out;

  const size_t S = (size_t)BN * KP;
  float* ws   = (float*)d_ws;
  float* mu   = ws;          float* sig  = ws + S;
  float* U    = ws + 2 * S;  float* W2   = ws + 3 * S;
  float* Y1   = ws + 4 * S;  float* bm   = ws + 5 * S;
  float* bY   = ws + 6 * S;  float* gobs = ws + 7 * S;
  float* nmu  = ws + 8 * S;  float* nsg  = ws + 9 * S;
  float* cvec = ws + 10 * S;
  float* dvec = cvec + BN;
  float* gsum = dvec + BN;

  k_init<<<BN, 256, 0, stream>>>(x, mu, sig);
  for (int step = 0; step < 20; ++step) {
    k_prep<<<BN, 256, 0, stream>>>(mu, sig, U, W2, Y1, cvec, dvec);
    k_attn<<<B_SZ * NT16, 128, 0, stream>>>(mu, U, W2, Y1, cvec, dvec, bm, bY);
    k_obs<<<B_SZ * NT16, 128, 0, stream>>>(mu, Wg, tgt, gobs);
    k_zero<<<1, 32, 0, stream>>>(gsum);
    k_gradnat<<<BN, 256, 0, stream>>>(mu, sig, bm, bY, gobs, pmu, psig, nmu,
                                      nsg, gsum);
    k_update<<<BN, 256, 0, stream>>>(mu, sig, nmu, nsg, gsum);
  }
  k_logits<<<dim3(B_SZ * NT16, V_SZ / 64), 128, 0, stream>>>(mu, Wg, out);
}